// ProteinAtomicEmbedder_20607253086910
// MI455X (gfx1250) — compile-verified
//
#include <hip/hip_runtime.h>
#include <math.h>

// ---------------------------------------------------------------------------
// ProteinAtomicEmbedder for MI455X (gfx1250, wave32, WMMA).
//
// Fused conv design: per 16-edge tile (one wave):
//   phase0: gather src node scalars/vectors + sh1, compute vdot, zero per-edge acc
//   phase1: hidden = ReLU(ef @ W1 + b1)           via v_wmma_f32_16x16x4_f32
//   phase2: for each 16-col chunk of w = h@W2+b2  via v_wmma_f32_16x16x4_f32,
//           immediately scatter chunk into per-edge accumulators (LDS ds_add_f32)
//   phase3: apply 1/sqrt2, 1/sqrt3 scales, global atomic add into segment sums
// This avoids materializing E x {400,800,1600} per-edge weights (~8 GB traffic).
// ---------------------------------------------------------------------------

typedef __attribute__((ext_vector_type(2))) float v2f;
typedef __attribute__((ext_vector_type(8))) float v8f;

namespace {

constexpr int kNA = 21000, kNR = 1500, kEA = 168000, kER = 45000;
constexpr int kNS = 16, kNV = 4, kRNS = 32, kRNV = 8;
constexpr int kAtomIn = 23, kNumAA = 21;
constexpr float kSQ3   = 1.7320508075688772f;
constexpr float kRSQ2  = 0.7071067811865476f;
constexpr float kRSQ3  = 0.5773502691896258f;

__device__ __forceinline__ v8f wmma4(v2f a, v2f b, v8f c) {
  // D(16x16,f32) = A(16x4,f32) * B(4x16,f32) + C
  return __builtin_amdgcn_wmma_f32_16x16x4_f32(
      false, a, false, b, (short)0, c, false, false);
}

// ---- streamed Linear-ReLU-Linear(-LayerNorm) helpers ----------------------

template <int DH, int DOUT, class F>
__device__ __forceinline__ void mlp_ln_stream(
    int DIN, F getx,
    const float* __restrict__ W1, const float* __restrict__ b1,
    const float* __restrict__ W2, const float* __restrict__ b2,
    const float* __restrict__ g,  const float* __restrict__ bb,
    float* __restrict__ out) {
  float h[DH];
#pragma unroll
  for (int j = 0; j < DH; ++j) h[j] = b1[j];
  for (int i = 0; i < DIN; ++i) {
    float xi = getx(i);
    const float* wr = W1 + i * DH;
#pragma unroll
    for (int j = 0; j < DH; ++j) h[j] += xi * wr[j];
  }
#pragma unroll
  for (int j = 0; j < DH; ++j) h[j] = fmaxf(h[j], 0.0f);
  float y[DOUT];
  float mu = 0.0f;
#pragma unroll
  for (int o = 0; o < DOUT; ++o) {
    float acc = b2[o];
#pragma unroll
    for (int j = 0; j < DH; ++j) acc += h[j] * W2[j * DOUT + o];
    y[o] = acc;
    mu += acc;
  }
  mu *= (1.0f / DOUT);
  float var = 0.0f;
#pragma unroll
  for (int o = 0; o < DOUT; ++o) {
    float d = y[o] - mu;
    var += d * d;
  }
  var *= (1.0f / DOUT);
  float rs = rsqrtf(var + 1e-5f);
#pragma unroll
  for (int o = 0; o < DOUT; ++o) out[o] = (y[o] - mu) * rs * g[o] + bb[o];
}

// ---- node embeddings -------------------------------------------------------

template <int DH, int DOUT>
__global__ void k_node_embed(int N, int DIN, const float* __restrict__ X,
                             const float* W1, const float* b1,
                             const float* W2, const float* b2,
                             const float* g, const float* bb,
                             float* __restrict__ out) {
  int n = blockIdx.x * blockDim.x + threadIdx.x;
  if (n >= N) return;
  const float* xr = X + n * DIN;
  mlp_ln_stream<DH, DOUT>(DIN, [&](int i) { return xr[i]; },
                          W1, b1, W2, b2, g, bb, out + n * DOUT);
}

// ---- edge feature kernels --------------------------------------------------

__global__ void k_atom_edge_feat(const float* __restrict__ coords,
                                 const float* __restrict__ feats,
                                 const int* __restrict__ dst,
                                 const int* __restrict__ src,
                                 const float* W1, const float* b1,
                                 const float* W2, const float* b2,
                                 const float* g, const float* bb,
                                 float* __restrict__ sh_out,
                                 float* __restrict__ ef_out) {
  int e = blockIdx.x * blockDim.x + threadIdx.x;
  if (e >= kEA) return;
  int d0 = dst[e], s0 = src[e];
  float vx = coords[d0 * 3 + 0] - coords[s0 * 3 + 0];
  float vy = coords[d0 * 3 + 1] - coords[s0 * 3 + 1];
  float vz = coords[d0 * 3 + 2] - coords[s0 * 3 + 2];
  float ax = vx + 1e-12f, ay = vy + 1e-12f, az = vz + 1e-12f;
  float dist = sqrtf(ax * ax + ay * ay + az * az);
  float inv = kSQ3 / dist;
  sh_out[e * 3 + 0] = vx * inv;
  sh_out[e * 3 + 1] = vy * inv;
  sh_out[e * 3 + 2] = vz * inv;
  const float invsig = 16.0f / 5.0f;
  const float* fd = feats + d0 * kAtomIn;
  const float* fs = feats + s0 * kAtomIn;
  auto getx = [&](int i) -> float {
    if (i < 23) return fd[i];
    if (i < 46) return fs[i - 23];
    int j = i - 46;
    float m = 5.0f * (float)j / 15.0f;
    float t = (dist - m) * invsig;
    return expf(-t * t);
  };
  mlp_ln_stream<16, 16>(62, getx, W1, b1, W2, b2, g, bb, ef_out + e * 16);
}

__global__ void k_agg_edge_feat(const float* __restrict__ acoords,
                                const float* __restrict__ rcoords,
                                const float* __restrict__ feats,
                                const int* __restrict__ batch,
                                const float* W1, const float* b1,
                                const float* W2, const float* b2,
                                const float* g, const float* bb,
                                float* __restrict__ sh_out,
                                float* __restrict__ ef_out) {
  int a = blockIdx.x * blockDim.x + threadIdx.x;
  if (a >= kNA) return;
  int r = batch[a];
  float vx = acoords[a * 3 + 0] - rcoords[r * 3 + 0];
  float vy = acoords[a * 3 + 1] - rcoords[r * 3 + 1];
  float vz = acoords[a * 3 + 2] - rcoords[r * 3 + 2];
  float ax = vx + 1e-12f, ay = vy + 1e-12f, az = vz + 1e-12f;
  float dist = sqrtf(ax * ax + ay * ay + az * az);
  float inv = kSQ3 / dist;
  sh_out[a * 3 + 0] = vx * inv;
  sh_out[a * 3 + 1] = vy * inv;
  sh_out[a * 3 + 2] = vz * inv;
  const float invsig = 16.0f / 10.0f;
  const float* fa = feats + a * kAtomIn;
  auto getx = [&](int i) -> float {
    if (i < 23) return fa[i];
    int j = i - 23;
    float m = 10.0f * (float)j / 15.0f;
    float t = (dist - m) * invsig;
    return expf(-t * t);
  };
  mlp_ln_stream<16, 16>(39, getx, W1, b1, W2, b2, g, bb, ef_out + a * 16);
}

__global__ void k_res_edge_feat(const float* __restrict__ coords,
                                const float* __restrict__ feats,
                                const int* __restrict__ dst,
                                const int* __restrict__ src,
                                const float* W1, const float* b1,
                                const float* W2, const float* b2,
                                const float* g, const float* bb,
                                float* __restrict__ sh_out,
                                float* __restrict__ ef_out) {
  int e = blockIdx.x * blockDim.x + threadIdx.x;
  if (e >= kER) return;
  int d0 = dst[e], s0 = src[e];
  float vx = coords[d0 * 3 + 0] - coords[s0 * 3 + 0];
  float vy = coords[d0 * 3 + 1] - coords[s0 * 3 + 1];
  float vz = coords[d0 * 3 + 2] - coords[s0 * 3 + 2];
  float ax = vx + 1e-12f, ay = vy + 1e-12f, az = vz + 1e-12f;
  float dist = sqrtf(ax * ax + ay * ay + az * az);
  float inv = kSQ3 / dist;
  sh_out[e * 3 + 0] = vx * inv;
  sh_out[e * 3 + 1] = vy * inv;
  sh_out[e * 3 + 2] = vz * inv;
  float delta = (float)(d0 - s0);
  const float invsig = 32.0f / 22.0f;
  const float kLogC = 0.5756462732485115f;  // ln(10000)/16
  const float* fd = feats + d0 * kNumAA;
  const float* fs = feats + s0 * kNumAA;
  auto getx = [&](int i) -> float {
    if (i < 21) return fd[i];
    if (i < 42) return fs[i - 21];
    if (i < 74) {
      int j = i - 42;
      float m = 22.0f * (float)j / 31.0f;
      float t = (dist - m) * invsig;
      return expf(-t * t);
    }
    int j = i - 74;
    if (j < 16) return sinf(delta * expf(-(float)j * kLogC));
    return cosf(delta * expf(-(float)(j - 16) * kLogC));
  };
  mlp_ln_stream<32, 32>(106, getx, W1, b1, W2, b2, g, bb, ef_out + e * 32);
}

// ---- utility kernels -------------------------------------------------------

__global__ void k_zero(float* __restrict__ p, int n) {
  int t = blockIdx.x * blockDim.x + threadIdx.x;
  if (t < n) p[t] = 0.0f;
}

__global__ void k_count(const int* __restrict__ idx, int n,
                        float* __restrict__ cnt) {
  int t = blockIdx.x * blockDim.x + threadIdx.x;
  if (t < n) atomicAdd(&cnt[idx[t]], 1.0f);
}

__global__ void k_finalize(int N, int DS, int DV3,
                           const float* __restrict__ accs,
                           const float* __restrict__ accv,
                           const float* __restrict__ cnt,
                           float* __restrict__ ns, float* __restrict__ nv) {
  int t = blockIdx.x * blockDim.x + threadIdx.x;
  int per = DS + DV3;
  if (t >= N * per) return;
  int n = t / per, j = t % per;
  float ci = 1.0f / fmaxf(cnt[n], 1.0f);
  if (j < DS)
    ns[n * DS + j] += accs[n * DS + j] * ci;
  else {
    int k = j - DS;
    nv[n * DV3 + k] += accv[n * DV3 + k] * ci;
  }
}

__global__ void k_output(const float* __restrict__ res_s,
                         const float* __restrict__ res_v,
                         float* __restrict__ out) {
  int t = blockIdx.x * blockDim.x + threadIdx.x;
  if (t >= kNR * 56) return;
  int n = t / 56, j = t % 56;
  out[t] = (j < 32) ? res_s[n * 32 + j] : res_v[n * 24 + (j - 32)];
}

// ---- fused WMMA conv -------------------------------------------------------
// One wave handles 16 edges; 4 waves per block (128 threads).

template <int NIS, int NIV, int NOS, int NOV, int HE, int HH, int WTOT>
__global__ __launch_bounds__(128) void k_conv(
    int E, const int* __restrict__ dst_idx, const int* __restrict__ src_idx,
    const float* __restrict__ node_s, const float* __restrict__ node_v,
    const float* __restrict__ ef, const float* __restrict__ shv,
    const float* __restrict__ W1, const float* __restrict__ b1,
    const float* __restrict__ W2, const float* __restrict__ b2,
    float* __restrict__ acc_s, float* __restrict__ acc_v) {
  constexpr int SEG_A = NIS * NOS;
  constexpr int SEG_B = SEG_A + NIV * NOS;
  constexpr int SEG_C = SEG_B + NIS * NOV;
  const int wid = threadIdx.x >> 5;
  const int lane = threadIdx.x & 31;
  const int base = (blockIdx.x * 4 + wid) * 16;
  const int half = lane >> 4;   // 0|1
  const int l16 = lane & 15;

  __shared__ float ls_ss[4][16][NIS];
  __shared__ float ls_sv[4][16][NIV * 3];
  __shared__ float ls_vd[4][16][NIV];
  __shared__ float ls_sh[4][16][4];
  __shared__ float ls_hid[4][16][HH];
  __shared__ float ls_es[4][16][NOS];
  __shared__ float ls_et[4][16][NOV];
  __shared__ float ls_ev[4][16][NOV * 3];

  // ---- phase 0: gather inputs, zero per-edge accumulators
  if (lane < 16) {
    int e = base + lane;
    int ec = (e < E) ? e : (E - 1);
    int s = src_idx ? src_idx[ec] : ec;
    const float* ssp = node_s + s * NIS;
#pragma unroll
    for (int i = 0; i < NIS; ++i) ls_ss[wid][lane][i] = ssp[i];
    float s0 = shv[ec * 3 + 0], s1 = shv[ec * 3 + 1], s2 = shv[ec * 3 + 2];
    ls_sh[wid][lane][0] = s0;
    ls_sh[wid][lane][1] = s1;
    ls_sh[wid][lane][2] = s2;
    const float* svp = node_v + s * NIV * 3;
#pragma unroll
    for (int v = 0; v < NIV; ++v) {
      float x = svp[v * 3 + 0], y = svp[v * 3 + 1], z = svp[v * 3 + 2];
      ls_sv[wid][lane][v * 3 + 0] = x;
      ls_sv[wid][lane][v * 3 + 1] = y;
      ls_sv[wid][lane][v * 3 + 2] = z;
      ls_vd[wid][lane][v] = (x * s0 + y * s1 + z * s2) * kRSQ3;
    }
  }
  for (int t = lane; t < 16 * NOS; t += 32) ls_es[wid][t / NOS][t % NOS] = 0.0f;
  for (int t = lane; t < 16 * NOV; t += 32) ls_et[wid][t / NOV][t % NOV] = 0.0f;
  for (int t = lane; t < 16 * NOV * 3; t += 32)
    ls_ev[wid][t / (NOV * 3)][t % (NOV * 3)] = 0.0f;
  __syncthreads();

  // ---- phase 1: hidden = ReLU(ef @ W1 + b1) via WMMA f32 16x16x4
  {
    int er = base + l16;
    er = (er < E) ? er : (E - 1);
    const float* efrow = ef + er * HE;
#pragma unroll
    for (int n0 = 0; n0 < HH; n0 += 16) {
      float bias = b1[n0 + l16];
      v8f c;
#pragma unroll
      for (int r = 0; r < 8; ++r) c[r] = bias;
#pragma unroll
      for (int k0 = 0; k0 < HE; k0 += 4) {
        v2f a, b;
        a[0] = efrow[k0 + 2 * half + 0];
        a[1] = efrow[k0 + 2 * half + 1];
        b[0] = W1[(k0 + 2 * half + 0) * HH + n0 + l16];
        b[1] = W1[(k0 + 2 * half + 1) * HH + n0 + l16];
        c = wmma4(a, b, c);
      }
#pragma unroll
      for (int r = 0; r < 8; ++r)
        ls_hid[wid][r + 8 * half][n0 + l16] = fmaxf(c[r], 0.0f);
    }
  }
  __syncthreads();

  // ---- phase 2: w = hidden @ W2 + b2, consumed chunk-by-chunk
  v2f afr[HH / 4];
#pragma unroll
  for (int kk = 0; kk < HH / 4; ++kk) {
    afr[kk][0] = ls_hid[wid][l16][kk * 4 + 2 * half + 0];
    afr[kk][1] = ls_hid[wid][l16][kk * 4 + 2 * half + 1];
  }
  for (int n0 = 0; n0 < WTOT; n0 += 16) {
    int idx = n0 + l16;
    float bias = b2[idx];
    v8f c;
#pragma unroll
    for (int r = 0; r < 8; ++r) c[r] = bias;
#pragma unroll
    for (int kk = 0; kk < HH / 4; ++kk) {
      v2f b;
      b[0] = W2[(kk * 4 + 2 * half + 0) * WTOT + idx];
      b[1] = W2[(kk * 4 + 2 * half + 1) * WTOT + idx];
      c = wmma4(afr[kk], b, c);
    }
    // lane holds column `idx` for edges {8*half .. 8*half+7}
    if (idx < SEG_A) {
      int i = idx / NOS, o = idx % NOS;
#pragma unroll
      for (int r = 0; r < 8; ++r) {
        int eL = r + 8 * half;
        atomicAdd(&ls_es[wid][eL][o], ls_ss[wid][eL][i] * c[r]);
      }
    } else if (idx < SEG_B) {
      int q = idx - SEG_A;
      int v = q / NOS, o = q % NOS;
#pragma unroll
      for (int r = 0; r < 8; ++r) {
        int eL = r + 8 * half;
        atomicAdd(&ls_es[wid][eL][o], ls_vd[wid][eL][v] * c[r]);
      }
    } else if (idx < SEG_C) {
      int q = idx - SEG_B;
      int i = q / NOV, o = q % NOV;
#pragma unroll
      for (int r = 0; r < 8; ++r) {
        int eL = r + 8 * half;
        atomicAdd(&ls_et[wid][eL][o], ls_ss[wid][eL][i] * c[r]);
      }
    } else {
      int q = idx - SEG_C;
      int v = q / NOV, o = q % NOV;
#pragma unroll
      for (int r = 0; r < 8; ++r) {
        int eL = r + 8 * half;
        float cv = c[r];
        atomicAdd(&ls_ev[wid][eL][o * 3 + 0], ls_sv[wid][eL][v * 3 + 0] * cv);
        atomicAdd(&ls_ev[wid][eL][o * 3 + 1], ls_sv[wid][eL][v * 3 + 1] * cv);
        atomicAdd(&ls_ev[wid][eL][o * 3 + 2], ls_sv[wid][eL][v * 3 + 2] * cv);
      }
    }
  }
  __syncthreads();

  // ---- phase 3: scale and scatter into segment sums
  constexpr int PERE = NOS + 3 * NOV;
  for (int t = lane; t < 16 * PERE; t += 32) {
    int eL = t / PERE, j = t % PERE;
    int e = base + eL;
    if (e < E) {
      int d = dst_idx[e];
      if (j < NOS) {
        atomicAdd(&acc_s[d * NOS + j], ls_es[wid][eL][j] * kRSQ2);
      } else {
        int o = (j - NOS) / 3, k = (j - NOS) % 3;
        float tmp = ls_et[wid][eL][o] * kRSQ3;
        float val = (tmp * ls_sh[wid][eL][k] + ls_ev[wid][eL][o * 3 + k]) * kRSQ2;
        atomicAdd(&acc_v[d * NOV * 3 + o * 3 + k], val);
      }
    }
  }
}

inline void zero_async(float* p, int n, hipStream_t s) {
  k_zero<<<(n + 255) / 256, 256, 0, s>>>(p, n);
}

}  // namespace

// ---------------------------------------------------------------------------

extern "C" void kernel_launch(void* const* d_in, const int* in_sizes, int n_in,
                              void* d_out, int out_size, void* d_ws,
                              size_t ws_size, hipStream_t stream) {
  (void)in_sizes; (void)n_in; (void)out_size; (void)ws_size;

  const float* atom_coords = (const float*)d_in[0];
  const float* atom_feats  = (const float*)d_in[1];
  const float* res_coords  = (const float*)d_in[2];
  const float* res_feats   = (const float*)d_in[3];
  const int* aei   = (const int*)d_in[4];
  const int* rei   = (const int*)d_in[5];
  const int* batch = (const int*)d_in[6];
  const int* a_dst = aei;       const int* a_src = aei + kEA;
  const int* r_dst = rei;       const int* r_src = rei + kER;

  auto F = [&](int i) { return (const float*)d_in[i]; };
  // params flattened as a JAX pytree (dict keys sorted at every level):
  //   agg_edge(7), atom_embed(13), layers(19 + 12*l), radius_edge(79),
  //   res_edge(85), res_embed(91)
  // mlp dict order: W1, W2, b, b1, b2, g ; ffn dict order: W1, W2, b1, b2
  // layer dict order: agg_ffn, atom_ffn, res_ffn
  const int I_AGG_E = 7, I_AT_EMB = 13, I_LAY = 19, I_RAD = 79, I_RES_E = 85,
            I_RES_EMB = 91;

  // workspace layout (floats)
  float* p = (float*)d_ws;
  float* a_sh      = p; p += kEA * 3;
  float* radius_ef = p; p += kEA * 16;
  float* g_sh      = p; p += kNA * 3;
  float* agg_ef    = p; p += kNA * 16;
  float* r_sh      = p; p += kER * 3;
  float* res_ef    = p; p += kER * 32;
  float* atom_s    = p; p += kNA * kNS;
  float* atom_v    = p; p += kNA * kNV * 3;
  float* res_s     = p; p += kNR * kRNS;
  float* res_v     = p; p += kNR * kRNV * 3;
  float* acc_as    = p; p += kNA * kNS;
  float* acc_av    = p; p += kNA * kNV * 3;
  float* acc_rs    = p; p += kNR * kRNS;
  float* acc_rv    = p; p += kNR * kRNV * 3;
  float* cnt_a     = p; p += kNA;
  float* cnt_g     = p; p += kNR;
  float* cnt_r     = p; p += kNR;

  // --- embeddings
  k_node_embed<16, 16><<<(kNA + 127) / 128, 128, 0, stream>>>(
      kNA, kAtomIn, atom_feats, F(I_AT_EMB + 0), F(I_AT_EMB + 3),
      F(I_AT_EMB + 1), F(I_AT_EMB + 4), F(I_AT_EMB + 5), F(I_AT_EMB + 2),
      atom_s);
  k_node_embed<32, 32><<<(kNR + 127) / 128, 128, 0, stream>>>(
      kNR, kNumAA, res_feats, F(I_RES_EMB + 0), F(I_RES_EMB + 3),
      F(I_RES_EMB + 1), F(I_RES_EMB + 4), F(I_RES_EMB + 5), F(I_RES_EMB + 2),
      res_s);
  zero_async(atom_v, kNA * kNV * 3, stream);
  zero_async(res_v, kNR * kRNV * 3, stream);

  // --- edge features (computed once, reused by every layer)
  k_atom_edge_feat<<<(kEA + 255) / 256, 256, 0, stream>>>(
      atom_coords, atom_feats, a_dst, a_src, F(I_RAD + 0), F(I_RAD + 3),
      F(I_RAD + 1), F(I_RAD + 4), F(I_RAD + 5), F(I_RAD + 2), a_sh, radius_ef);
  k_agg_edge_feat<<<(kNA + 255) / 256, 256, 0, stream>>>(
      atom_coords, res_coords, atom_feats, batch, F(I_AGG_E + 0),
      F(I_AGG_E + 3), F(I_AGG_E + 1), F(I_AGG_E + 4), F(I_AGG_E + 5),
      F(I_AGG_E + 2), g_sh, agg_ef);
  k_res_edge_feat<<<(kER + 255) / 256, 256, 0, stream>>>(
      res_coords, res_feats, r_dst, r_src, F(I_RES_E + 0), F(I_RES_E + 3),
      F(I_RES_E + 1), F(I_RES_E + 4), F(I_RES_E + 5), F(I_RES_E + 2), r_sh,
      res_ef);

  // --- segment counts (graph-constant)
  zero_async(cnt_a, kNA, stream);
  zero_async(cnt_g, kNR, stream);
  zero_async(cnt_r, kNR, stream);
  k_count<<<(kEA + 255) / 256, 256, 0, stream>>>(a_dst, kEA, cnt_a);
  k_count<<<(kNA + 255) / 256, 256, 0, stream>>>(batch, kNA, cnt_g);
  k_count<<<(kER + 255) / 256, 256, 0, stream>>>(r_dst, kER, cnt_r);

  // --- message-passing layers
  for (int l = 0; l < 5; ++l) {
    const int LB = I_LAY + 12 * l;
    const float* aggW1 = F(LB + 0), *aggW2 = F(LB + 1), *aggB1 = F(LB + 2),
                *aggB2 = F(LB + 3);
    const float* atW1 = F(LB + 4), *atW2 = F(LB + 5), *atB1 = F(LB + 6),
                *atB2 = F(LB + 7);
    const float* rsW1 = F(LB + 8), *rsW2 = F(LB + 9), *rsB1 = F(LB + 10),
                *rsB2 = F(LB + 11);

    // atom <- atom conv
    zero_async(acc_as, kNA * kNS, stream);
    zero_async(acc_av, kNA * kNV * 3, stream);
    k_conv<16, 4, 16, 4, 16, 32, 400><<<(kEA + 63) / 64, 128, 0, stream>>>(
        kEA, a_dst, a_src, atom_s, atom_v, radius_ef, a_sh, atW1, atB1, atW2,
        atB2, acc_as, acc_av);
    k_finalize<<<(kNA * 28 + 255) / 256, 256, 0, stream>>>(
        kNA, kNS, kNV * 3, acc_as, acc_av, cnt_a, atom_s, atom_v);

    // res <- atom aggregation conv (src = identity over atoms)
    zero_async(acc_rs, kNR * kRNS, stream);
    zero_async(acc_rv, kNR * kRNV * 3, stream);
    k_conv<16, 4, 32, 8, 16, 32, 800><<<(kNA + 63) / 64, 128, 0, stream>>>(
        kNA, batch, (const int*)nullptr, atom_s, atom_v, agg_ef, g_sh, aggW1,
        aggB1, aggW2, aggB2, acc_rs, acc_rv);
    k_finalize<<<(kNR * 56 + 255) / 256, 256, 0, stream>>>(
        kNR, kRNS, kRNV * 3, acc_rs, acc_rv, cnt_g, res_s, res_v);

    // res <- res conv
    zero_async(acc_rs, kNR * kRNS, stream);
    zero_async(acc_rv, kNR * kRNV * 3, stream);
    k_conv<32, 8, 32, 8, 32, 64, 1600><<<(kER + 63) / 64, 128, 0, stream>>>(
        kER, r_dst, r_src, res_s, res_v, res_ef, r_sh, rsW1, rsB1, rsW2, rsB2,
        acc_rs, acc_rv);
    k_finalize<<<(kNR * 56 + 255) / 256, 256, 0, stream>>>(
        kNR, kRNS, kRNV * 3, acc_rs, acc_rv, cnt_r, res_s, res_v);
  }

  // --- output: concat [res_s (NRx32), res_v (NRx24)]
  k_output<<<(kNR * 56 + 255) / 256, 256, 0, stream>>>(res_s, res_v,
                                                       (float*)d_out);
}